// Qwen3MoeGate_63952063038000
// MI455X (gfx1250) — compile-verified
//
#include <hip/hip_runtime.h>
#include <hip/hip_bf16.h>
#include <math.h>
#include <stdint.h>

typedef float v2f __attribute__((ext_vector_type(2)));
typedef float v8f __attribute__((ext_vector_type(8)));

#define HDIM     2048
#define NEXPERT  128
#define TOPK     8
#define KCHUNK   64
#define NCHUNK   (HDIM / KCHUNK)        // 32
#define LDS_STR  (KCHUNK + 4)           // 68 floats = 272 B: 16B-aligned rows
                                        // (async B128 needs 16B LDS alignment),
                                        // banks (4r + c) % 64 conflict-free.

// One 256-thread block (8 wave32) handles 16 tokens x 128 experts.
// Wave w computes the 16x16 logit tile for experts [16w, 16w+16) via
// v_wmma_f32_16x16x4_f32. The A tile (16 tokens x 64 k) is double-buffered in
// LDS and filled with GLOBAL_LOAD_ASYNC_TO_LDS_B128 (ASYNCcnt), overlapping
// the next chunk's HBM fetch with the current chunk's 16 WMMAs. B rows
// (weight, 1 MB total) stay L2-resident and are loaded per-wave.
__global__ __launch_bounds__(256) void moe_gate_kernel(
    const float* __restrict__ x,        // [T, H] row-major
    const float* __restrict__ w,        // [E, H] row-major
    float* __restrict__ out_idx,        // [T, 8] indices as float
    float* __restrict__ out_wt)         // [T, 8] renormalized weights
{
    __shared__ float lds_x[2][16 * LDS_STR];         // double-buffered A tile
    __shared__ float lds_logits[16 * NEXPERT];       // 16 tokens x 128 experts

    const int tok0 = blockIdx.x * 16;
    const int tid  = threadIdx.x;
    const int wave = tid >> 5;           // 0..7 -> expert tile
    const int lane = tid & 31;
    const int half = lane >> 4;          // 0: K pair {0,1}; 1: K pair {2,3}
    const int l16  = lane & 15;

    const int expert0 = wave * 16;
    // B-matrix: lane holds column N = l16 (expert row of W), K pair by half.
    const float* wrow = w + (size_t)(expert0 + l16) * HDIM;

    // Per-thread async-staging coordinates: 256 threads x 16B covers the
    // 16x64-float chunk exactly.
    const int si   = tid << 2;           // 0..1023 (floats)
    const int srow = si >> 6;
    const int scol = si & 63;
    const float* gsrc = x + (size_t)(tok0 + srow) * HDIM + scol;
    const unsigned ldsdst0 = (unsigned)(uintptr_t)&lds_x[0][srow * LDS_STR + scol];
    const unsigned ldsdst1 = (unsigned)(uintptr_t)&lds_x[1][srow * LDS_STR + scol];

    // Preload chunk 0 into buffer 0 (memory -> LDS DMA, no VGPR round trip).
    asm volatile("global_load_async_to_lds_b128 %0, %1, off"
                 :: "v"(ldsdst0), "v"(gsrc) : "memory");

    v8f acc = {};

    for (int c = 0; c < NCHUNK; ++c) {
        const int k0 = c * KCHUNK;
        if (c + 1 < NCHUNK) {
            // Issue chunk c+1 into the other buffer (last read two barriers
            // ago, so it is safe to overwrite), then wait for chunk c only:
            // async loads complete in order, so asynccnt<=1 means chunk c
            // has landed in LDS.
            const unsigned dst = ((c + 1) & 1) ? ldsdst1 : ldsdst0;
            asm volatile("global_load_async_to_lds_b128 %0, %1, off"
                         :: "v"(dst), "v"(gsrc + k0 + KCHUNK) : "memory");
            asm volatile("s_wait_asynccnt 0x1" ::: "memory");
        } else {
            asm volatile("s_wait_asynccnt 0x0" ::: "memory");
        }
        __syncthreads();   // every wave's DMA for chunk c is complete

        const float* lx = &lds_x[c & 1][0];
        #pragma unroll
        for (int kk = 0; kk < KCHUNK; kk += 4) {
            const int ka = kk + half * 2;
            // A 16x4 layout: lanes 0-15 rows M=0..15 (K=0 in v0, K=1 in v1),
            // lanes 16-31 same rows (K=2 in v0, K=3 in v1).
            v2f a;
            a.x = lx[l16 * LDS_STR + ka];
            a.y = lx[l16 * LDS_STR + ka + 1];
            // B 4x16 layout mirrors A: v0 = K={0|2}, v1 = K={1|3}, N = l16.
            const float2 bv = *(const float2*)(wrow + k0 + ka);
            v2f b; b.x = bv.x; b.y = bv.y;
            acc = __builtin_amdgcn_wmma_f32_16x16x4_f32(
                false, a, false, b, (short)0, acc, false, false);
        }
        __syncthreads();   // all waves done reading buffer c&1
    }

    // C/D layout: VGPR r -> token r (lanes 0-15) / token r+8 (lanes 16-31),
    // expert = expert0 + l16.
    #pragma unroll
    for (int r = 0; r < 8; ++r)
        lds_logits[(r + 8 * half) * NEXPERT + expert0 + l16] = acc[r];
    __syncthreads();

    // Epilogue: one thread per token. Top-8 of logits == top-8 of softmax
    // scores (monotone), and renormalized weights = softmax over the 8
    // selected logits (the full-softmax normalizer cancels).
    if (tid < 16) {
        float* lg = &lds_logits[tid * NEXPERT];
        int   idxs[TOPK];
        float vals[TOPK];
        #pragma unroll
        for (int j = 0; j < TOPK; ++j) {
            float best = -INFINITY;
            int   bi   = 0;
            for (int e = 0; e < NEXPERT; ++e) {
                const float v = lg[e];
                if (v > best) { best = v; bi = e; }   // first index wins ties
            }
            lg[bi]  = -INFINITY;   // exclusive row: knock out the winner
            idxs[j] = bi;
            vals[j] = best;
        }
        const float m = vals[0];
        float ex[TOPK];
        float s = 0.f;
        #pragma unroll
        for (int j = 0; j < TOPK; ++j) { ex[j] = __expf(vals[j] - m); s += ex[j]; }
        const float inv = 1.f / s;
        const size_t o = (size_t)(tok0 + tid) * TOPK;
        #pragma unroll
        for (int j = 0; j < TOPK; ++j) {
            out_idx[o + j] = (float)idxs[j];
            out_wt[o + j]  = ex[j] * inv;
        }
    }
}

extern "C" void kernel_launch(void* const* d_in, const int* in_sizes, int n_in,
                              void* d_out, int out_size, void* d_ws, size_t ws_size,
                              hipStream_t stream) {
    const float* x = (const float*)d_in[0];   // [T, 2048]
    const float* w = (const float*)d_in[1];   // [128, 2048]
    const int T = in_sizes[0] / HDIM;         // 16384

    float* out_idx = (float*)d_out;           // [T, 8] indices (as float)
    float* out_wt  = (float*)d_out + (size_t)T * TOPK; // [T, 8] weights

    const int nblocks = T / 16;               // 1024
    moe_gate_kernel<<<nblocks, 256, 0, stream>>>(x, w, out_idx, out_wt);
}